// PMLoss_7868380086324
// MI455X (gfx1250) — compile-verified
//
#include <hip/hip_runtime.h>
#include <math.h>

typedef __attribute__((ext_vector_type(2))) float v2f;
typedef __attribute__((ext_vector_type(8))) float v8f;

#define B_ROIS 128
#define C_CLS  22
#define P_PTS  1024

__device__ __forceinline__ void quat2mat(float w, float x, float y, float z, float R[9]) {
    R[0] = 1.f - 2.f*(y*y + z*z); R[1] = 2.f*(x*y - w*z);       R[2] = 2.f*(x*z + w*y);
    R[3] = 2.f*(x*y + w*z);       R[4] = 1.f - 2.f*(x*x + z*z); R[5] = 2.f*(y*z - w*x);
    R[6] = 2.f*(x*z - w*y);       R[7] = 2.f*(y*z + w*x);       R[8] = 1.f - 2.f*(x*x + y*y);
}

__global__ void pml_init(float* out) { out[0] = 0.0f; }

__global__ __launch_bounds__(256) void pml_kernel(
    const float* __restrict__ prediction,  // [B, 4C]
    const float* __restrict__ target,      // [B, 4C]
    const float* __restrict__ weight,      // [B, 4C]
    const float* __restrict__ points,      // [C, P, 3]
    const float* __restrict__ symmetry,    // [C]
    float* __restrict__ out)               // [1]
{
    __shared__ float sA[P_PTS * 4];   // per point p: (-2ppx, -2ppy, -2ppz, 1)
    __shared__ float sB[P_PTS * 4];   // per point q: (ptx, pty, ptz, |pt|^2)
    __shared__ float sPP[P_PTS];      // |pp|^2
    __shared__ float sSum;

    const int b   = blockIdx.x;
    const int tid = threadIdx.x;
    if (tid == 0) sSum = 0.0f;

    // ---- active class: argmax over weight[b, c, 0] ----
    int   cls = 0;
    float bw  = weight[b * 4 * C_CLS + 0];
    for (int c = 1; c < C_CLS; ++c) {
        float w = weight[b * 4 * C_CLS + c * 4];
        if (w > bw) { bw = w; cls = c; }
    }
    const bool isSym = symmetry[cls] > 0.0f;

    // ---- quaternions -> rotation matrices (redundant per thread, uniform) ----
    const int qoff = b * 4 * C_CLS + cls * 4;
    float qw = prediction[qoff + 0], qx = prediction[qoff + 1];
    float qy = prediction[qoff + 2], qz = prediction[qoff + 3];
    float inv = rsqrtf(fmaxf(qw*qw + qx*qx + qy*qy + qz*qz, 1e-12f));
    qw *= inv; qx *= inv; qy *= inv; qz *= inv;
    float tw = target[qoff + 0], tx = target[qoff + 1];
    float ty = target[qoff + 2], tz = target[qoff + 3];

    float Rp[9], Rt[9];
    quat2mat(qw, qx, qy, qz, Rp);
    quat2mat(tw, tx, ty, tz, Rt);

    // ---- rotate points, stage WMMA fragments in LDS ----
    float local_pw = 0.0f;
    for (int p = tid; p < P_PTS; p += blockDim.x) {
        const int po = (cls * P_PTS + p) * 3;
        float px = points[po + 0], py = points[po + 1], pz = points[po + 2];

        float ppx = Rp[0]*px + Rp[1]*py + Rp[2]*pz;
        float ppy = Rp[3]*px + Rp[4]*py + Rp[5]*pz;
        float ppz = Rp[6]*px + Rp[7]*py + Rp[8]*pz;
        float ptx = Rt[0]*px + Rt[1]*py + Rt[2]*pz;
        float pty = Rt[3]*px + Rt[4]*py + Rt[5]*pz;
        float ptz = Rt[6]*px + Rt[7]*py + Rt[8]*pz;

        sA[p*4 + 0] = -2.0f * ppx;
        sA[p*4 + 1] = -2.0f * ppy;
        sA[p*4 + 2] = -2.0f * ppz;
        sA[p*4 + 3] = 1.0f;
        sB[p*4 + 0] = ptx;
        sB[p*4 + 1] = pty;
        sB[p*4 + 2] = ptz;
        sB[p*4 + 3] = ptx*ptx + pty*pty + ptz*ptz;
        sPP[p]      = ppx*ppx + ppy*ppy + ppz*ppz;

        float dx = ppx - ptx, dy = ppy - pty, dz = ppz - ptz;
        local_pw += dx*dx + dy*dy + dz*dz;
    }
    __syncthreads();  // LDS fragments ready; also orders sSum=0 before atomics

    if (!isSym) {
        // non-symmetric: pointwise squared distance
        atomicAdd(&sSum, local_pw);
    } else {
        // symmetric: d_sym[p] = |pp[p]|^2 + min_q ( -2 pp.pt + |pt[q]|^2 ) via WMMA
        const int lane   = tid & 31;
        const int wave   = tid >> 5;
        const int nwaves = blockDim.x >> 5;       // 8
        const int m      = lane & 15;             // row (A) / col (B) within tile
        const int kb     = (lane >> 4) << 1;      // K base: 0 for lanes 0-15, 2 for 16-31

        float local_sym = 0.0f;
        for (int tp = wave; tp < P_PTS / 16; tp += nwaves) {
            v2f a;
            a.x = sA[(tp*16 + m)*4 + kb + 0];
            a.y = sA[(tp*16 + m)*4 + kb + 1];

            float mn[8];
            #pragma unroll
            for (int i = 0; i < 8; ++i) mn[i] = 3.4e38f;

            for (int tq = 0; tq < P_PTS / 16; ++tq) {
                v2f bf;
                bf.x = sB[(tq*16 + m)*4 + kb + 0];
                bf.y = sB[(tq*16 + m)*4 + kb + 1];
                v8f c = {};
                // D = A(16x4) x B(4x16) : one 16x16 tile of (-2 pp.pt + |pt|^2)
                v8f d = __builtin_amdgcn_wmma_f32_16x16x4_f32(
                    /*neg_a=*/false, a, /*neg_b=*/false, bf,
                    /*c_mod=*/(short)0, c, /*reuse_a=*/false, /*reuse_b=*/false);
                #pragma unroll
                for (int i = 0; i < 8; ++i) mn[i] = fminf(mn[i], d[i]);
            }

            // min over the 16 columns held across lanes {0-15} / {16-31}
            #pragma unroll
            for (int i = 0; i < 8; ++i) {
                float v = mn[i];
                v = fminf(v, __shfl_xor(v, 1, 32));
                v = fminf(v, __shfl_xor(v, 2, 32));
                v = fminf(v, __shfl_xor(v, 4, 32));
                v = fminf(v, __shfl_xor(v, 8, 32));
                mn[i] = v;
            }
            // lane 0 owns rows M=0..7 (p = tp*16+i); lane 16 owns rows M=8..15
            if (m == 0) {
                const int base = tp*16 + ((lane >> 4) << 3);
                #pragma unroll
                for (int i = 0; i < 8; ++i)
                    local_sym += sPP[base + i] + mn[i];
            }
        }
        atomicAdd(&sSum, local_sym);
    }

    __syncthreads();
    if (tid == 0)
        atomicAdd(out, sSum * (1.0f / (2.0f * B_ROIS * P_PTS)));
}

extern "C" void kernel_launch(void* const* d_in, const int* in_sizes, int n_in,
                              void* d_out, int out_size, void* d_ws, size_t ws_size,
                              hipStream_t stream) {
    const float* prediction = (const float*)d_in[0];
    const float* target     = (const float*)d_in[1];
    const float* weight     = (const float*)d_in[2];
    const float* points     = (const float*)d_in[3];
    const float* symmetry   = (const float*)d_in[4];
    float* out = (float*)d_out;

    pml_init<<<1, 1, 0, stream>>>(out);
    pml_kernel<<<B_ROIS, 256, 0, stream>>>(prediction, target, weight, points, symmetry, out);
}